// MCLLayer_80461917323853
// MI455X (gfx1250) — compile-verified
//
#include <hip/hip_runtime.h>
#include <hip/hip_bf16.h>
#include <math.h>

typedef float v2f __attribute__((ext_vector_type(2)));
typedef float v8f __attribute__((ext_vector_type(8)));

#define GAMMA   20.0f
#define GAMMA2  10.0f
#define KATZ    0.5f
#define KATZ2   0.25f   // KATZ^2

// ---- problem constants (from setup_inputs) ----
#define NB   16
#define NS   25     // support shots total
#define NC   640
#define HW   25
#define NQ   150
#define NWAY 5
#define KSH  5
#define MS   125    // n_way*HW
#define MSP  128    // padded cols for S
#define TBS  130    // row stride for Tsq/Bm (bank-conflict-free b64 frags)
#define QTS  642    // row stride for transposed query tile [m][c]
#define NINST (NB*NQ)   // 2400

// ---------------------------------------------------------------------------
// Kernel 1: support prototype mean + L2 norm over c, stored as B-matrix
// layout Bmat[b][c][nj] with nj padded to 128 (pad cols zeroed).
// grid = NB*NWAY*HW = 2000 blocks, 128 threads.
// ---------------------------------------------------------------------------
__global__ void prep_support_kernel(const float* __restrict__ sup,
                                    float* __restrict__ Bmat) {
    const int bi  = blockIdx.x;
    const int b   = bi / (NWAY * HW);
    const int nj  = bi % (NWAY * HW);
    const int n   = nj / HW;
    const int j   = nj % HW;
    const int tid = threadIdx.x;

    __shared__ float red[128];

    float val[5];
    float ss = 0.0f;
#pragma unroll
    for (int i = 0; i < 5; ++i) {
        const int c = tid + i * 128;
        float acc = 0.0f;
#pragma unroll
        for (int k = 0; k < KSH; ++k) {
            const size_t idx =
                ((((size_t)b * NS + (size_t)(n * KSH + k)) * NC + c) * HW) + j;
            acc += sup[idx];
        }
        acc *= (1.0f / (float)KSH);
        val[i] = acc;
        ss += acc * acc;
    }
    red[tid] = ss;
    __syncthreads();
    for (int ofs = 64; ofs > 0; ofs >>= 1) {
        if (tid < ofs) red[tid] += red[tid + ofs];
        __syncthreads();
    }
    const float inv = 1.0f / (1e-16f + sqrtf(red[0]));

#pragma unroll
    for (int i = 0; i < 5; ++i) {
        const int c = tid + i * 128;
        Bmat[(((size_t)b * NC) + c) * MSP + nj] = val[i] * inv;
    }
    // zero padding columns 125..127 (done once per b by the nj==0 block)
    if (nj == 0) {
        for (int idx = tid; idx < NC * (MSP - MS); idx += 128) {
            const int c = idx / (MSP - MS);
            const int p = MS + idx % (MSP - MS);
            Bmat[(((size_t)b * NC) + c) * MSP + p] = 0.0f;
        }
    }
}

// ---------------------------------------------------------------------------
// Kernel 2: one block per (b,q) instance. 256 threads = 8 waves.
//   Phase A: load query tile transposed to LDS [m][c] (stride 642),
//            L2-normalize per spatial pos.
//   Phase B: S(25x125) = q_n^T * sup_n via V_WMMA_F32_16X16X4_F32
//            (M-tiles at row offsets {0, 9} -> no predication; A frags are
//             contiguous ds_load_b64, bank-conflict-free).
//   Phase C: softmaxes; G = Bm * Tsq^T (25x25, K=128) again via WMMA with
//            tiles at offsets {0,9}; Gauss-Jordan solve; class scores.
// ---------------------------------------------------------------------------
// LDS layout (floats). [0,16050) is the phase-A query tile, aliased by all
// post-GEMM buffers. Phase-A-live scratch lives at >= 16050.
#define OFF_QT   0        // 25*642 = 16050 (phase A + main GEMM only)
#define OFF_S    0        // 25*128 -> reserve 4096
#define OFF_TSQ  4096     // 25*130 = 3250
#define OFF_BM   7346     // 25*130 = 3250
#define OFF_U    10596    // 128
#define OFF_MAUG 10724    // 25*26 = 650
#define OFF_PIV  11374    // 26
#define OFF_FV   11400    // 25
#define OFF_VQ   11425    // 25
#define OFF_PART 16050    // 200 (phase A partial sums)
#define OFF_SSQ  16250    // 25  (phase A inverse norms)
#define SMEM_FLOATS 16278 // 65112 bytes

__global__ __launch_bounds__(256)
void mcl_main_kernel(const float* __restrict__ query,
                     const float* __restrict__ Bmat,
                     float* __restrict__ out) {
    __shared__ float smem[SMEM_FLOATS];

    const int tid  = threadIdx.x;
    const int inst = blockIdx.x;          // b*NQ + q
    const int b    = inst / NQ;

    // ------- Phase A: transpose-load query tile + L2 normalize rows -------
    const float* qsrc = query + (size_t)inst * (NC * HW);
    for (int i = tid; i < NC * HW; i += 256) {
        const int c = i / HW;
        const int m = i % HW;
        smem[OFF_QT + m * QTS + c] = qsrc[i];   // coalesced read, xposed write
    }
    __syncthreads();

    if (tid < 200) {                       // 8 partial sums per spatial pos
        const int m  = tid >> 3;
        const int c0 = tid & 7;
        float s = 0.0f;
        for (int c = c0; c < NC; c += 8) {
            const float x = smem[OFF_QT + m * QTS + c];
            s += x * x;
        }
        smem[OFF_PART + tid] = s;
    }
    __syncthreads();
    if (tid < HW) {
        float s = 0.0f;
#pragma unroll
        for (int i = 0; i < 8; ++i) s += smem[OFF_PART + tid * 8 + i];
        smem[OFF_SSQ + tid] = 1.0f / (1e-16f + sqrtf(s));
    }
    __syncthreads();
    for (int i = tid; i < NC * HW; i += 256) {
        const int m = i % HW;
        smem[OFF_QT + m * QTS + (i / HW)] *= smem[OFF_SSQ + m];
    }
    __syncthreads();

    // ---------------- Phase B: S = A * B via f32 WMMA ----------------
    // wave w owns N-tile w (cols w*16..w*16+15); M-tiles at rows {0,9}.
    const int w    = tid >> 5;
    const int lane = tid & 31;
    const int mrow = lane & 15;       // M (and N) index within tile
    const int kh   = lane >> 4;       // K half selector
    const int col  = w * 16 + mrow;   // global column 0..127

    const float* Bb = Bmat + (size_t)b * NC * MSP;

    v8f c0 = {};
    v8f c1 = {};
#pragma unroll 4
    for (int k = 0; k < NC; k += 4) {
        const int k0 = k + 2 * kh;    // even -> 8B-aligned b64 LDS loads
        const v2f a0 = *reinterpret_cast<const v2f*>(
            &smem[OFF_QT + mrow * QTS + k0]);
        const v2f a1 = *reinterpret_cast<const v2f*>(
            &smem[OFF_QT + (mrow + 9) * QTS + k0]);
        v2f bf;
        bf.x = Bb[(size_t)k0 * MSP + col];
        bf.y = Bb[(size_t)(k0 + 1) * MSP + col];
        c0 = __builtin_amdgcn_wmma_f32_16x16x4_f32(false, a0, false, bf,
                                                   (short)0, c0, false, false);
        c1 = __builtin_amdgcn_wmma_f32_16x16x4_f32(false, a1, false, bf,
                                                   (short)0, c1, false, false);
    }
    __syncthreads();   // qt no longer needed; S region may alias it

    // C/D layout: VGPR v, lanes 0-15 -> M=v, lanes 16-31 -> M=v+8
#pragma unroll
    for (int v = 0; v < 8; ++v) {
        const int r0 = v + 8 * kh;
        smem[OFF_S + r0 * MSP + col]       = c0[v];   // rows 0..15
        smem[OFF_S + (9 + r0) * MSP + col] = c1[v];   // rows 9..24 (dups identical)
    }
    __syncthreads();

    // ---------------- Phase C1: softmaxes (+ zero pad cols of Tsq/Bm) -----
    if (tid < HW) {                       // rows: T_sq, gamma=20, over 125
        const int m = tid;
        float mx = -1e30f;
        for (int k = 0; k < MS; ++k)
            mx = fmaxf(mx, smem[OFF_S + m * MSP + k]);
        float s = 0.0f;
        for (int k = 0; k < MS; ++k) {
            const float e = __expf(GAMMA * (smem[OFF_S + m * MSP + k] - mx));
            smem[OFF_TSQ + m * TBS + k] = e;
            s += e;
        }
        const float inv = 1.0f / s;
        for (int k = 0; k < MS; ++k) smem[OFF_TSQ + m * TBS + k] *= inv;
    } else if (tid >= 32 && tid < 32 + MS) {   // cols: B = T_qs^T, gamma=10
        const int k = tid - 32;
        float mx = -1e30f;
        for (int m = 0; m < HW; ++m)
            mx = fmaxf(mx, smem[OFF_S + m * MSP + k]);
        float s = 0.0f;
        for (int m = 0; m < HW; ++m) {
            const float e = __expf(GAMMA2 * (smem[OFF_S + m * MSP + k] - mx));
            smem[OFF_BM + m * TBS + k] = e;
            s += e;
        }
        const float inv = 1.0f / s;
        for (int m = 0; m < HW; ++m) smem[OFF_BM + m * TBS + k] *= inv;
    } else if (tid >= 160 && tid < 160 + HW * 3) {  // zero K-pad cols 125..127
        const int idx = tid - 160;
        const int m = idx / 3;
        const int k = MS + idx % 3;
        smem[OFF_TSQ + m * TBS + k] = 0.0f;
        smem[OFF_BM  + m * TBS + k] = 0.0f;
    }
    __syncthreads();

    // ---------------- Phase C2: Maug = [I - c^2*Bm*Tsq^T | 1 + c*Bm*1] ----
    // G(25x25) via WMMA: waves 0..3 own (mt,nt) tiles at offsets {0,9};
    // wave 7 computes the rhs column.
    if (w < 4) {
        const int ma = (w >> 1) * 9;
        const int na = (w & 1) * 9;
        v8f g = {};
#pragma unroll 4
        for (int k = 0; k < MSP; k += 4) {
            const int k0 = k + 2 * kh;   // even -> aligned b64 LDS loads
            const v2f a = *reinterpret_cast<const v2f*>(
                &smem[OFF_BM + (ma + mrow) * TBS + k0]);
            const v2f bf = *reinterpret_cast<const v2f*>(
                &smem[OFF_TSQ + (na + mrow) * TBS + k0]);
            g = __builtin_amdgcn_wmma_f32_16x16x4_f32(false, a, false, bf,
                                                      (short)0, g, false, false);
        }
#pragma unroll
        for (int v = 0; v < 8; ++v) {
            const int r  = ma + v + 8 * kh;       // 0..24
            const int cc = na + mrow;             // 0..24
            smem[OFF_MAUG + r * 26 + cc] =
                ((r == cc) ? 1.0f : 0.0f) - KATZ2 * g[v];
        }
    } else if (w == 7 && lane < HW) {
        const int r = lane;
        float s = 0.0f;
        for (int k = 0; k < MS; ++k) s += smem[OFF_BM + r * TBS + k];
        smem[OFF_MAUG + r * 26 + 25] = 1.0f + KATZ * s;
    }
    __syncthreads();

    // ---------------- Phase C3: Gauss-Jordan (column diag-dominant) -------
    for (int i = 0; i < 25; ++i) {
        if (tid < 26) smem[OFF_PIV + tid] = smem[OFF_MAUG + i * 26 + tid];
        __syncthreads();
        if (tid < 25)
            smem[OFF_FV + tid] =
                smem[OFF_MAUG + tid * 26 + i] / smem[OFF_PIV + i];
        __syncthreads();
        for (int idx = tid; idx < 25 * 26; idx += 256) {
            const int r = idx / 26;
            if (r != i)
                smem[OFF_MAUG + idx] -=
                    smem[OFF_FV + r] * smem[OFF_PIV + (idx % 26)];
        }
        __syncthreads();
    }
    if (tid < 25)
        smem[OFF_VQ + tid] =
            smem[OFF_MAUG + tid * 26 + 25] / smem[OFF_MAUG + tid * 26 + tid];
    __syncthreads();

    // ---------------- Phase C4: u = Tsq^T v_q ; normalize; class sums -----
    if (tid < MS) {
        float s = 0.0f;
        for (int m = 0; m < HW; ++m)
            s += smem[OFF_TSQ + m * TBS + tid] * smem[OFF_VQ + m];
        smem[OFF_U + tid] = s;
    }
    __syncthreads();
    if (tid == 0) {
        float s = 0.0f;
        for (int k = 0; k < MS; ++k) s += smem[OFF_U + k];
        smem[OFF_U + 127] = 1.0f / s;
    }
    __syncthreads();
    if (tid < NWAY) {
        float s = 0.0f;
        for (int j = 0; j < HW; ++j) s += smem[OFF_U + tid * HW + j];
        out[(size_t)inst * NWAY + tid] = s * smem[OFF_U + 127];
    }
}

// ---------------------------------------------------------------------------
extern "C" void kernel_launch(void* const* d_in, const int* in_sizes, int n_in,
                              void* d_out, int out_size, void* d_ws, size_t ws_size,
                              hipStream_t stream) {
    (void)in_sizes; (void)n_in; (void)out_size; (void)ws_size;
    const float* support = (const float*)d_in[0];   // (16,25,640,5,5) f32
    const float* query   = (const float*)d_in[2];   // (16,150,640,5,5) f32
    float* Bmat = (float*)d_ws;                     // 16*640*128 f32 = 5.24 MB
    float* out  = (float*)d_out;                    // (2400, 5) f32

    prep_support_kernel<<<NB * NWAY * HW, 128, 0, stream>>>(support, Bmat);
    mcl_main_kernel<<<NINST, 256, 0, stream>>>(query, Bmat, out);
}